// GCNConv_8907762172421
// MI455X (gfx1250) — compile-verified
//
#include <hip/hip_runtime.h>
#include <hip/hip_bf16.h>

typedef __attribute__((ext_vector_type(2))) float v2f;
typedef __attribute__((ext_vector_type(8))) float v8f;

// ---------------------------------------------------------------------------
// Kernel 1: out[N,64] = node_states[N,64] @ W[64,64] + b[64]
// One wave32 computes one 16x16 output tile via V_WMMA_F32_16X16X4_F32,
// iterating K = 0..63 in steps of 4.
//
// VGPR layouts (CDNA5 ISA 7.12.2, 32-bit):
//   A 16x4 : lane<16  -> row M=lane,    v0=A[M][k0+0], v1=A[M][k0+1]
//            lane>=16 -> row M=lane-16, v0=A[M][k0+2], v1=A[M][k0+3]
//   B 4x16 : lane<16  -> col N=lane,    v0=B[k0+0][N], v1=B[k0+1][N]
//            lane>=16 -> col N=lane-16, v0=B[k0+2][N], v1=B[k0+3][N]
//   C/D    : 8 VGPRs, element (M = r + 8*(lane/16), N = lane%16)
// ---------------------------------------------------------------------------
__global__ __launch_bounds__(256) void gcn_gemm_bias_kernel(
    const float* __restrict__ A,    // [Nrows, 64] node_states
    const float* __restrict__ W,    // [64, 64]
    const float* __restrict__ bias, // [64]
    float* __restrict__ out,        // [Nrows, 64]
    int Nrows)
{
    const int wave_in_block = threadIdx.x >> 5;   // 0..7
    const int lane          = threadIdx.x & 31;
    const int row_tile      = wave_in_block >> 2; // 0..1 (two 16-row tiles per block)
    const int col_tile      = wave_in_block & 3;  // 0..3 (four 16-col tiles)

    const int row_base = (blockIdx.x * 2 + row_tile) * 16;
    if (row_base >= Nrows) return;                // wave-uniform: EXEC stays all-1s

    const int n_base = col_tile * 16;
    const int half   = lane >> 4;                 // 0: K pair {0,1}, 1: K pair {2,3}
    const int m      = lane & 15;                 // row (A) / column (B,C,D) index

    const float* arow = A + (long long)(row_base + m) * 64;

    v8f acc = {};
    #pragma unroll
    for (int k0 = 0; k0 < 64; k0 += 4) {
        const int ka = k0 + 2 * half;
        v2f a;
        a.x = arow[ka + 0];                       // contiguous float2 per lane
        a.y = arow[ka + 1];
        v2f b;
        b.x = W[(ka + 0) * 64 + n_base + m];      // column of W, strided
        b.y = W[(ka + 1) * 64 + n_base + m];
        // 8 args: (neg_a, A, neg_b, B, c_mod, C, reuse_a, reuse_b)
        acc = __builtin_amdgcn_wmma_f32_16x16x4_f32(
            false, a, false, b, (short)0, acc, false, false);
    }

    const float bv = bias[n_base + m];
    #pragma unroll
    for (int r = 0; r < 8; ++r) {
        const int row = row_base + r + 8 * half;
        out[(long long)row * 64 + n_base + m] = acc[r] + bv;
    }
}

// ---------------------------------------------------------------------------
// Kernel 2: for each edge e: out[dst[e], :] += edge_weight[e] * node_states[src[e], :]
// One wave per edge; lane covers 2 columns (float2 gather, 2 f32 atomics).
// node_states + out are L2-resident (51 MB << 192 MB L2), so gathers hit L2
// and the atomic adds resolve at the L2 atomic units.
// ---------------------------------------------------------------------------
__global__ __launch_bounds__(256) void gcn_edge_scatter_kernel(
    const float* __restrict__ node_states, // [N, 64]
    const float* __restrict__ edge_weight, // [E]
    const int*   __restrict__ src_index,   // [E]
    const int*   __restrict__ dst_index,   // [E]
    float* __restrict__ out,               // [N, 64]
    long long E)
{
    const long long gtid = (long long)blockIdx.x * blockDim.x + threadIdx.x;
    const long long e    = gtid >> 5;             // edge id (one wave per edge)
    if (e >= E) return;                           // wave-uniform exit
    const int lane = threadIdx.x & 31;

    const int   s = src_index[e];
    const int   d = dst_index[e];
    const float w = edge_weight[e];

    const float2* srow = (const float2*)(node_states + (long long)s * 64);
    float2 v = srow[lane];                        // lanes 0..31 cover 64 floats

    float* drow = out + (long long)d * 64 + lane * 2;
    atomicAdd(drow + 0, v.x * w);                 // global_atomic_add_f32 (no return)
    atomicAdd(drow + 1, v.y * w);
}

extern "C" void kernel_launch(void* const* d_in, const int* in_sizes, int n_in,
                              void* d_out, int out_size, void* d_ws, size_t ws_size,
                              hipStream_t stream) {
    const float* node_states = (const float*)d_in[0]; // [N, 64]
    const float* edge_weight = (const float*)d_in[1]; // [E]
    const float* W           = (const float*)d_in[2]; // [64, 64]
    const float* b           = (const float*)d_in[3]; // [64]
    const int*   src_index   = (const int*)d_in[4];   // [E]
    const int*   dst_index   = (const int*)d_in[5];   // [E]
    float*       out         = (float*)d_out;         // [N, 64]

    const int       Nrows = in_sizes[0] / 64;         // 100000
    const long long E     = in_sizes[1];              // 3200000

    // Kernel 1: overwrite d_out with node_states @ W + b.
    // 8 waves/block -> 32 rows x 64 cols per block.
    const int gemm_blocks = (Nrows + 31) / 32;
    gcn_gemm_bias_kernel<<<gemm_blocks, 256, 0, stream>>>(
        node_states, W, b, out, Nrows);

    // Kernel 2: scatter-add edge messages on top (same stream -> ordered).
    const long long edges_per_block = 8;              // 8 waves of 32
    const long long scat_blocks = (E + edges_per_block - 1) / edges_per_block;
    gcn_edge_scatter_kernel<<<(unsigned)scat_blocks, 256, 0, stream>>>(
        node_states, edge_weight, src_index, dst_index, out, E);
}